// PairwiseSiteInteraction_5617817223844
// MI455X (gfx1250) — compile-verified
//
#include <hip/hip_runtime.h>
#include <hip/hip_bf16.h>
#include <math.h>
#include <stdint.h>

typedef float v2f  __attribute__((ext_vector_type(2)));
typedef float v8f  __attribute__((ext_vector_type(8)));
typedef unsigned int u32x4 __attribute__((ext_vector_type(4)));
typedef int   i32x4 __attribute__((ext_vector_type(4)));
typedef int   i32x8 __attribute__((ext_vector_type(8)));

#define BLOCK 256
#define STAGE 2048                 // edges staged per TDM round (per array)
#define STAGES_PER_BLOCK 8
#define CHUNK (STAGE * STAGES_PER_BLOCK)   // 16384 edges per block
#define MAX_SEG 128

// ---------------------------------------------------------------------------
// TDM: 1-D tensor DMA of `tile_elems` int32 from global to LDS.
// D# built per CDNA5 ISA (group0/group1 packing).
// tensor_dim0 = remaining valid elements -> OOB tail reads return zero.
// 6-arg builtin variant (clang-23 / therock-10.0 headers).
// ---------------------------------------------------------------------------
__device__ __forceinline__ void tdm_load_1d(unsigned lds_off, const void* gptr,
                                            unsigned valid_elems, unsigned tile_elems) {
  unsigned long long ga = (unsigned long long)gptr;
  u32x4 g0;
  g0.x = 0x1u;                                        // count=1, user mode, no gather
  g0.y = lds_off;                                     // lds_addr (bytes)
  g0.z = (unsigned)(ga & 0xFFFFFFFFu);                // global_addr[31:0]
  g0.w = (unsigned)((ga >> 32) & 0x01FFFFFFu)         // global_addr[56:32]
       | (2u << 30);                                  // type = 2 ("image")
  i32x8 g1;
  g1[0] = (int)(2u << 16);                            // data_size = 4 bytes; mask=0
  g1[1] = (int)((valid_elems & 0xFFFFu) << 16);       // tensor_dim0[15:0]
  g1[2] = (int)((valid_elems >> 16) & 0xFFFFu);       // tensor_dim0[31:16]; tensor_dim1=0
  g1[3] = (int)((tile_elems & 0xFFFFu) << 16);        // tile_dim0
  g1[4] = 0;                                          // tile_dim1/2 unused
  g1[5] = (int)valid_elems;                           // tensor_dim0_stride (unused, 1 row)
  g1[6] = 0;
  g1[7] = 0;
  i32x4 z4 = {0, 0, 0, 0};
  i32x8 z8 = {0, 0, 0, 0, 0, 0, 0, 0};
  __builtin_amdgcn_tensor_load_to_lds(g0, g1, z4, z4, z8, 0);
}

// ---------------------------------------------------------------------------
// Main kernel: TDM double-buffered edge streaming + L2-resident gathers +
// sorted-run segment accumulation into LDS buckets. Each block emits its 64
// per-segment partial sums into d_ws (no global atomics).
// ---------------------------------------------------------------------------
__global__ void __launch_bounds__(BLOCK) lj_main_kernel(
    const float* __restrict__ pos,      // [N,3]
    const float* __restrict__ par,      // [N,2]
    const int*   __restrict__ src,      // [E]
    const int*   __restrict__ dst,      // [E]
    const int*   __restrict__ bat,      // [E] sorted
    float*       __restrict__ partials, // [nBlocks, nSeg]
    int nEdges, int nSeg)
{
  __shared__ int   s_src[2][STAGE];
  __shared__ int   s_dst[2][STAGE];
  __shared__ int   s_bat[2][STAGE];
  __shared__ float s_acc[MAX_SEG];

  const int t = threadIdx.x;
  for (int i = t; i < nSeg; i += BLOCK) s_acc[i] = 0.0f;

  const long long blockBase = (long long)blockIdx.x * CHUNK;
  long long remEdges = (long long)nEdges - blockBase;
  int myStages = (int)((remEdges + STAGE - 1) / STAGE);
  if (myStages > STAGES_PER_BLOCK) myStages = STAGES_PER_BLOCK;

  const bool issuer = (t < 32);   // wave 0 drives the TDM pipeline

  if (issuer) {
    unsigned valid = (unsigned)remEdges;   // OOB tail handled by tensor_dim0 check
    tdm_load_1d((unsigned)(size_t)&s_src[0][0], src + blockBase, valid, STAGE);
    tdm_load_1d((unsigned)(size_t)&s_dst[0][0], dst + blockBase, valid, STAGE);
    tdm_load_1d((unsigned)(size_t)&s_bat[0][0], bat + blockBase, valid, STAGE);
  }

  int   curB = -1;
  float acc  = 0.0f;

  for (int s = 0; s < myStages; ++s) {
    if (issuer) {
      if (s + 1 < myStages) {
        long long nbase = blockBase + (long long)(s + 1) * STAGE;
        unsigned nvalid = (unsigned)((long long)nEdges - nbase);
        int nbuf = (s + 1) & 1;
        tdm_load_1d((unsigned)(size_t)&s_src[nbuf][0], src + nbase, nvalid, STAGE);
        tdm_load_1d((unsigned)(size_t)&s_dst[nbuf][0], dst + nbase, nvalid, STAGE);
        tdm_load_1d((unsigned)(size_t)&s_bat[nbuf][0], bat + nbase, nvalid, STAGE);
        __builtin_amdgcn_s_wait_tensorcnt(3);   // stage s's 3 DMAs complete
      } else {
        __builtin_amdgcn_s_wait_tensorcnt(0);
      }
    }
    __syncthreads();                             // stage s visible to all waves

    const int buf = s & 1;
    const long long stageBase = blockBase + (long long)s * STAGE;
#pragma unroll
    for (int j = 0; j < STAGE / BLOCK; ++j) {
      const int k = t + j * BLOCK;
      const long long i = stageBase + k;
      if (i < (long long)nEdges) {
        const int si = s_src[buf][k];
        const int di = s_dst[buf][k];
        const int b  = s_bat[buf][k];
        // gathers: 4 MB site tables live in L2
        const float dx = pos[3 * si + 0] - pos[3 * di + 0];
        const float dy = pos[3 * si + 1] - pos[3 * di + 1];
        const float dz = pos[3 * si + 2] - pos[3 * di + 2];
        const float r2 = dx * dx + dy * dy + dz * dz;
        const float2 ps = ((const float2*)par)[si];
        const float2 pd = ((const float2*)par)[di];
        const float sg = 0.5f * (ps.x + pd.x);            // Lorentz
        const float ep = sqrtf(ps.y * pd.y);              // Berthelot
        const float x2 = (sg * sg) / r2;                  // (sigma/r)^2
        const float x6 = x2 * x2 * x2;
        const float e  = 4.0f * ep * (x6 * x6 - x6);
        if (b != curB) {                                   // sorted: rare flush
          if (curB >= 0) atomicAdd(&s_acc[curB], acc);     // ds_add_f32
          curB = b; acc = 0.0f;
        }
        acc += e;
      }
    }
    __syncthreads();   // all waves done with buf before it is overwritten
  }

  if (curB >= 0) atomicAdd(&s_acc[curB], acc);
  __syncthreads();

  for (int i = t; i < nSeg; i += BLOCK)
    partials[(long long)blockIdx.x * nSeg + i] = s_acc[i];
}

// ---------------------------------------------------------------------------
// Final reduction: out[n] = sum_r partials[r][n] via V_WMMA_F32_16X16X4_F32.
// A = ones(16x4) -> D[m][n] = sum_k B[k][n]; accumulate 4 rows per WMMA into C.
// With all-ones A the result is invariant to B's K-slot layout; every row of
// C equals the column-sum vector, so row 0 (lanes 0..15, VGPR 0) is the answer.
// Each wave owns 16 columns; one 128-thread block covers 64 segments.
// ---------------------------------------------------------------------------
__global__ void __launch_bounds__(128) seg_reduce_wmma_kernel(
    const float* __restrict__ P, float* __restrict__ out, int nRows, int nSeg)
{
  const int wave = threadIdx.x >> 5;
  const int lane = threadIdx.x & 31;
  const int col  = blockIdx.x * 64 + wave * 16 + (lane & 15);
  const int half = lane >> 4;            // which pair of K rows this lane feeds
  const bool colOK = (col < nSeg);

  v8f c = {0.f, 0.f, 0.f, 0.f, 0.f, 0.f, 0.f, 0.f};
  const v2f a = {1.0f, 1.0f};            // all-ones A matrix

  for (int r = 0; r < nRows; r += 4) {   // uniform trip count: EXEC all-1 at WMMA
    const int r0 = r + half * 2;
    const float b0 = (colOK && (r0     < nRows)) ? P[(long long)r0       * nSeg + col] : 0.0f;
    const float b1 = (colOK && (r0 + 1 < nRows)) ? P[(long long)(r0 + 1) * nSeg + col] : 0.0f;
    v2f b = {b0, b1};
    c = __builtin_amdgcn_wmma_f32_16x16x4_f32(
        /*neg_a=*/false, a, /*neg_b=*/false, b,
        /*c_mod=*/(short)0, c, /*reuse_a=*/false, /*reuse_b=*/false);
  }

  if (lane < 16 && colOK) out[col] = c[0];   // (M=0, N=lane)
}

// ---------------------------------------------------------------------------
extern "C" void kernel_launch(void* const* d_in, const int* in_sizes, int n_in,
                              void* d_out, int out_size, void* d_ws, size_t ws_size,
                              hipStream_t stream) {
  const float* pos   = (const float*)d_in[0];   // [N,3] f32
  const float* par   = (const float*)d_in[1];   // [N,2] f32
  const int*   eidx  = (const int*)d_in[2];     // [2,E] int32
  const int*   batch = (const int*)d_in[3];     // [E]   int32 (sorted)

  const int nEdges = in_sizes[3];
  const int nSeg   = out_size;                  // 64 graphs
  const int* src = eidx;
  const int* dst = eidx + nEdges;

  float* partials = (float*)d_ws;               // [nBlocks, nSeg] f32
  const int nBlocks = (nEdges + CHUNK - 1) / CHUNK;

  lj_main_kernel<<<nBlocks, BLOCK, 0, stream>>>(pos, par, src, dst, batch,
                                                partials, nEdges, nSeg);

  const int rBlocks = (nSeg + 63) / 64;
  seg_reduce_wmma_kernel<<<rBlocks, 128, 0, stream>>>(partials, (float*)d_out,
                                                      nBlocks, nSeg);
}